// SimplexTreeModel_63642825392498
// MI455X (gfx1250) — compile-verified
//
#include <hip/hip_runtime.h>

// Reference: out = F[inds].reshape(50, 2)  -> 100 gathered float32 values.
// F: 30M floats (d_in[0]), inds: 100 int32 (d_in[1]), d_out: 100 floats.
//
// Latency-bound micro-gather. Strategy:
//  - single wave32 (native CDNA5 width), one block, minimal launch overhead
//  - each lane owns 4 consecutive output elements:
//      * one coalesced global_load_b128 for 4 indices
//      * 4 independent scattered gathers issued before the wait (MLP)
//      * one coalesced global_store_b128 for 4 results
//  - non-temporal TH hints on every VMEM op: F is 120MB touched at 100 random
//    words with zero reuse -> do not pollute WGP$/L2 (gfx1250 th:TH_LOAD_NT).

// __builtin_nontemporal_* requires true clang vector types, not HIP's
// HIP_vector_type structs -> use ext_vector_type typedefs.
typedef int   v4i __attribute__((ext_vector_type(4)));
typedef float v4f __attribute__((ext_vector_type(4)));

#define TH_CARD 50
#define TH_OUT  (2 * TH_CARD)   // 100 elements
#define TH_PER_LANE 4           // 100 / 4 = 25 active lanes in one wave32

__global__ void simplex_gather_kernel(const float* __restrict__ F,
                                      const int*   __restrict__ inds,
                                      float*       __restrict__ out) {
    const int base = (int)threadIdx.x * TH_PER_LANE;
    if (base < TH_OUT) {
        // One b128 load of 4 consecutive indices (streaming, non-temporal).
        const v4i idx = __builtin_nontemporal_load(
            reinterpret_cast<const v4i*>(inds + base));

        // Four independent scattered gathers: all four global_load_b32
        // th:TH_LOAD_NT issue before a single s_wait_loadcnt, so the HBM
        // latency of all lanes' gathers overlaps (100 loads in flight).
        const float r0 = __builtin_nontemporal_load(F + idx.x);
        const float r1 = __builtin_nontemporal_load(F + idx.y);
        const float r2 = __builtin_nontemporal_load(F + idx.z);
        const float r3 = __builtin_nontemporal_load(F + idx.w);

        // One b128 coalesced store of the 4 results.
        v4f v;
        v.x = r0; v.y = r1; v.z = r2; v.w = r3;
        __builtin_nontemporal_store(v, reinterpret_cast<v4f*>(out + base));
    }
}

extern "C" void kernel_launch(void* const* d_in, const int* in_sizes, int n_in,
                              void* d_out, int out_size, void* d_ws, size_t ws_size,
                              hipStream_t stream) {
    (void)in_sizes; (void)n_in; (void)d_ws; (void)ws_size; (void)out_size;

    const float* F    = reinterpret_cast<const float*>(d_in[0]);
    const int*   inds = reinterpret_cast<const int*>(d_in[1]);
    float*       out  = reinterpret_cast<float*>(d_out);

    // 100 elements, 4 per lane -> 25 active lanes: exactly one wave32.
    simplex_gather_kernel<<<1, 32, 0, stream>>>(F, inds, out);
}